// SpanClassifier_17377437680220
// MI455X (gfx1250) — compile-verified
//
#include <hip/hip_runtime.h>

#define BB 64
#define SS 512
#define HH 768
#define LL 20
#define MS 8
#define NSP 9  // MAX_SPANS+1

typedef __attribute__((ext_vector_type(2))) float v2f;
typedef __attribute__((ext_vector_type(8))) float v8f;

// int workspace layout (element offsets into (int*)d_ws)
#define WS_N    0
#define WS_CNT  (BB)
#define WS_OS   (2*BB)
#define WS_OE   (2*BB + BB*MS)
#define WS_IS0  (2*BB + 2*BB*MS)
#define WS_IE0  (2*BB + 2*BB*MS + BB)

__device__ __forceinline__ float wave_sum(float v) {
#pragma unroll
  for (int off = 16; off > 0; off >>= 1) v += __shfl_xor(v, off, 32);
  return v;
}

// ---------------- Kernel 1: fused start/end logits (bandwidth pass) ----------------
// one wave per 4 rows of tokens_rep; weights register-blocked (row-invariant per lane)
// block = 256 threads = 8 waves -> 32 rows per block
__global__ void k_row_logits(const float* __restrict__ tok,
                             const float* __restrict__ wsv,
                             const float* __restrict__ wev,
                             const float* __restrict__ bsp,
                             const float* __restrict__ bep,
                             float* __restrict__ so, float* __restrict__ eo) {
  int wave = threadIdx.x >> 5;
  int lane = threadIdx.x & 31;
  int row0 = (blockIdx.x * 8 + wave) * 4;  // [0, BB*SS) in steps of 4
  float4 wsr[6], wer[6];
#pragma unroll
  for (int i = 0; i < 6; ++i) {
    int idx = (i * 32 + lane) * 4;
    wsr[i] = *(const float4*)(wsv + idx);
    wer[i] = *(const float4*)(wev + idx);
  }
  float bs = bsp[0], be = bep[0];
#pragma unroll
  for (int r = 0; r < 4; ++r) {
    const float* t = tok + (size_t)(row0 + r) * HH;
    float s = 0.f, e = 0.f;
#pragma unroll
    for (int i = 0; i < 6; ++i) {
      float4 tv = *(const float4*)(t + (i * 32 + lane) * 4);
      s += tv.x * wsr[i].x + tv.y * wsr[i].y + tv.z * wsr[i].z + tv.w * wsr[i].w;
      e += tv.x * wer[i].x + tv.y * wer[i].y + tv.z * wer[i].z + tv.w * wer[i].w;
    }
    s = wave_sum(s);
    e = wave_sum(e);
    if (lane == 0) {
      so[row0 + r] = s + bs;
      eo[row0 + r] = e + be;
    }
  }
}

// ---------------- Kernel 2: seq-level logits (fsl + n_spans) ----------------
__global__ void k_seq(const float* __restrict__ seq,
                      const float* __restrict__ W_span, const float* __restrict__ b_span,
                      const float* __restrict__ W_nsp, const float* __restrict__ b_nsp,
                      float* __restrict__ fsl_out, float* __restrict__ nsp_out,
                      int* __restrict__ ws_i) {
  __shared__ float res[LL + NSP];
  int b = blockIdx.x;
  int wave = threadIdx.x >> 5, lane = threadIdx.x & 31;
  const float* s = seq + b * HH;
  for (int j = wave; j < LL + NSP; j += 8) {
    float acc = 0.f;
    if (j < LL) {
      for (int k = lane; k < HH; k += 32) acc += s[k] * W_span[k * LL + j];
    } else {
      int jj = j - LL;
      for (int k = lane; k < HH; k += 32) acc += s[k] * W_nsp[k * NSP + jj];
    }
    acc = wave_sum(acc);
    if (lane == 0) res[j] = acc + (j < LL ? b_span[j] : b_nsp[j - LL]);
  }
  __syncthreads();
  if (threadIdx.x < LL) fsl_out[b * LL + threadIdx.x] = res[threadIdx.x];
  if (threadIdx.x < NSP) nsp_out[b * NSP + threadIdx.x] = res[LL + threadIdx.x];
  if (threadIdx.x == 0) {
    float best = res[LL];
    int bi = 0;
    for (int j = 1; j < NSP; ++j)
      if (res[LL + j] > best) { best = res[LL + j]; bi = j; }
    ws_i[WS_N + b] = bi;
  }
}

// ---------------- Kernel 3: per-item top-8 + sort + span selection ----------------
// one wave per item
__global__ void k_select(const float* __restrict__ start_logits,
                         const float* __restrict__ end_logits,
                         int* __restrict__ ws_i) {
  int b = blockIdx.x, lane = threadIdx.x;
  __shared__ float buf[SS];
  __shared__ int topi_s[MS], topi_e[MS];
  for (int pass = 0; pass < 2; ++pass) {
    const float* src = (pass == 0 ? start_logits : end_logits) + b * SS;
    int* out = (pass == 0) ? topi_s : topi_e;
    for (int i = lane; i < SS; i += 32) buf[i] = src[i];
    __syncthreads();
    for (int r = 0; r < MS; ++r) {
      float best = -3.0e38f;
      int bi = 0x7fffffff;
      for (int i = lane; i < SS; i += 32) {
        float v = buf[i];
        if (v > best) { best = v; bi = i; }  // ascending scan: first max kept
      }
#pragma unroll
      for (int off = 16; off > 0; off >>= 1) {
        float ov = __shfl_xor(best, off, 32);
        int oi = __shfl_xor(bi, off, 32);
        if (ov > best || (ov == best && oi < bi)) { best = ov; bi = oi; }
      }
      if (lane == 0) { out[r] = bi; buf[bi] = -3.0e38f; }
      __syncthreads();
    }
    __syncthreads();
  }
  if (lane == 0) {
    int n = ws_i[WS_N + b];
    int ss[MS], es[MS];
    for (int r = 0; r < MS; ++r) {
      ss[r] = (r < n) ? topi_s[r] : SS;  // sentinel SS sorts last
      es[r] = (r < n) ? topi_e[r] : SS;
    }
    for (int i = 1; i < MS; ++i) { int key = ss[i], j = i - 1; while (j >= 0 && ss[j] > key) { ss[j + 1] = ss[j]; --j; } ss[j + 1] = key; }
    for (int i = 1; i < MS; ++i) { int key = es[i], j = i - 1; while (j >= 0 && es[j] > key) { es[j + 1] = es[j]; --j; } es[j + 1] = key; }
    int si = 0, ei = 0, last = 0, rem = n, cnt = 0;
    int os[MS], oe[MS];
    for (int r = 0; r < MS; ++r) { os[r] = 0; oe[r] = 0; }
    for (int it = 0; it < 2 * MS; ++it) {
      bool active = (rem > 0) && (si < n) && (ei < n);
      int s = ss[si < MS - 1 ? si : MS - 1];
      int e = es[ei < MS - 1 ? ei : MS - 1];
      bool accept = active && (e > s) && (s >= last);
      bool ince = active && !accept && (e <= s);
      bool incs = active && !accept && (e > s);
      if (accept) { os[cnt] = s; oe[cnt] = e; last = e; ++cnt; --rem; ++si; ++ei; }
      else { si += incs ? 1 : 0; ei += ince ? 1 : 0; }
    }
    ws_i[WS_CNT + b] = cnt;
    for (int r = 0; r < MS; ++r) { ws_i[WS_OS + b * MS + r] = os[r]; ws_i[WS_OE + b * MS + r] = oe[r]; }
    ws_i[WS_IS0 + b] = topi_s[0];
    ws_i[WS_IE0 + b] = topi_e[0];
  }
}

// ---------------- Kernel 4: span pooling + WMMA span classifier + outputs ----------------
__global__ void k_span_pool(const float* __restrict__ tok,
                            const float* __restrict__ W_span, const float* __restrict__ b_span,
                            const float* __restrict__ fslp,  // staged full_seq_logits
                            const int* __restrict__ ws_i,
                            float* __restrict__ spans_out, float* __restrict__ starts_oh,
                            float* __restrict__ ends_oh, float* __restrict__ labels_oh) {
  __shared__ float sv[MS][HH];    // span mean vectors
  __shared__ float mlog[MS][32];  // raw WMMA logits (32 cols, first LL used)
  __shared__ float mls[MS][LL];   // softmaxed
  __shared__ float fsoft[LL];
  __shared__ int sh_os[MS], sh_oe[MS], sh_lab[MS];
  __shared__ int sh_cnt, sh_n, sh_is0, sh_ie0, sh_fslarg;
  int b = blockIdx.x, tid = threadIdx.x;
  if (tid == 0) {
    sh_cnt = ws_i[WS_CNT + b];
    sh_n = ws_i[WS_N + b];
    sh_is0 = ws_i[WS_IS0 + b];
    sh_ie0 = ws_i[WS_IE0 + b];
  }
  if (tid < MS) { sh_os[tid] = ws_i[WS_OS + b * MS + tid]; sh_oe[tid] = ws_i[WS_OE + b * MS + tid]; }
  __syncthreads();

  // span mean-pooling (reads hit L2 after kernel 1's pass)
  for (int idx = tid; idx < MS * HH; idx += 256) {
    int k = idx / HH, h = idx - k * HH;
    int s0 = sh_os[k], e0 = sh_oe[k];
    const float* tp = tok + ((size_t)b * SS + s0) * HH + h;
    float acc = 0.f;
    for (int t = s0; t <= e0; ++t) { acc += *tp; tp += HH; }
    sv[k][h] = acc / (float)(e0 - s0 + 1);
  }
  __syncthreads();

  // WMMA (wave 0, EXEC all-1s): [16x768] x [768x32] via f32 16x16x4, two N tiles
  if (tid < 32) {
    int lane = tid;
    int m = lane & 15;             // A row / B col (local)
    int kh = (lane >> 4) << 1;     // K half offset
    float amask = (m < MS) ? 1.f : 0.f;
    int n1 = m + 16;
    int n1c = (n1 < LL) ? n1 : 0;
    float bmask1 = (n1 < LL) ? 1.f : 0.f;
    v8f acc0 = {0.f, 0.f, 0.f, 0.f, 0.f, 0.f, 0.f, 0.f};
    v8f acc1 = {0.f, 0.f, 0.f, 0.f, 0.f, 0.f, 0.f, 0.f};
    for (int k0 = 0; k0 < HH; k0 += 4) {
      int ka = k0 + kh;
      v2f a, b0, b1;
      a.x = sv[m & (MS - 1)][ka] * amask;
      a.y = sv[m & (MS - 1)][ka + 1] * amask;
      b0.x = W_span[ka * LL + m];
      b0.y = W_span[(ka + 1) * LL + m];
      b1.x = W_span[ka * LL + n1c] * bmask1;
      b1.y = W_span[(ka + 1) * LL + n1c] * bmask1;
      acc0 = __builtin_amdgcn_wmma_f32_16x16x4_f32(false, a, false, b0, (short)0, acc0, false, false);
      acc1 = __builtin_amdgcn_wmma_f32_16x16x4_f32(false, a, false, b1, (short)0, acc1, false, false);
    }
    if (lane < 16) {  // VGPR r, lanes 0-15 -> row r (rows 0..7 are the real spans)
#pragma unroll
      for (int r = 0; r < 8; ++r) {
        mlog[r][lane] = acc0[r];
        mlog[r][lane + 16] = acc1[r];
      }
    }
  }
  __syncthreads();

  // per-row softmax + argmax label
  if (tid < MS) {
    int r = tid;
    float mx = -3.0e38f;
    float tmp[LL];
    for (int j = 0; j < LL; ++j) { float v = mlog[r][j] + b_span[j]; tmp[j] = v; if (v > mx) mx = v; }
    float sum = 0.f;
    for (int j = 0; j < LL; ++j) { float v = __expf(tmp[j] - mx); tmp[j] = v; sum += v; }
    float inv = 1.f / sum;
    float best = -1.f;
    int bi = 0;
    for (int j = 0; j < LL; ++j) { float v = tmp[j] * inv; mls[r][j] = v; if (v > best) { best = v; bi = j; } }
    sh_lab[r] = bi;
  }
  if (tid == 32) {  // fsl softmax + argmax (different wave than rows)
    float mx = -3.0e38f;
    float tmp[LL];
    for (int j = 0; j < LL; ++j) { float v = fslp[b * LL + j]; tmp[j] = v; if (v > mx) mx = v; }
    float sum = 0.f;
    for (int j = 0; j < LL; ++j) { float v = __expf(tmp[j] - mx); tmp[j] = v; sum += v; }
    float inv = 1.f / sum;
    float best = -1.f;
    int bi = 0;
    for (int j = 0; j < LL; ++j) { float v = tmp[j] * inv; fsoft[j] = v; if (v > best) { best = v; bi = j; } }
    sh_fslarg = bi;
  }
  __syncthreads();

  bool single = (sh_n <= 1);
  bool hasv = (sh_cnt > 0);
  if (tid < LL) {
    int j = tid;
    float spv;
    if (single) spv = fsoft[j];
    else if (!hasv) spv = 0.f;
    else { float bv = -3.0e38f; for (int r = 0; r < sh_cnt; ++r) bv = fmaxf(bv, mls[r][j]); spv = bv; }
    spans_out[b * LL + j] = spv;
    float lab;
    if (single || !hasv) lab = (j == sh_fslarg) ? 1.f : 0.f;
    else { lab = 0.f; for (int r = 0; r < sh_cnt; ++r) if (sh_lab[r] == j) lab = 1.f; }
    labels_oh[b * LL + j] = lab;
  }
  for (int t = tid; t < SS; t += 256) { starts_oh[b * SS + t] = 0.f; ends_oh[b * SS + t] = 0.f; }
  __syncthreads();
  if (tid == 0) {
    if (single) { starts_oh[b * SS + sh_is0] = 1.f; ends_oh[b * SS + sh_ie0] = 1.f; }
    else {
      for (int r = 0; r < sh_cnt; ++r) { starts_oh[b * SS + sh_os[r]] = 1.f; ends_oh[b * SS + sh_oe[r]] = 1.f; }
    }
  }
}

// ---------------- Kernel 5: pol_logits via f32 WMMA ----------------
// one block, 8 waves; wave = (mt in 0..3, nt in 0..1); K = 788 = 4*197
__global__ void k_pol(const float* __restrict__ seq, const float* __restrict__ spans,
                      const float* __restrict__ W_pol, const float* __restrict__ b_pol,
                      float* __restrict__ pol_out) {
  int wave = threadIdx.x >> 5, lane = threadIdx.x & 31;
  int mt = wave >> 1, nt = wave & 1;
  int m = mt * 16 + (lane & 15);
  int kh = (lane >> 4) << 1;
  int nl = lane & 15;
  int n = nt * 16 + nl;
  int nc = (n < LL) ? n : 0;
  float bmask = (n < LL) ? 1.f : 0.f;
  v8f acc = {0.f, 0.f, 0.f, 0.f, 0.f, 0.f, 0.f, 0.f};
  for (int k0 = 0; k0 < HH + LL; k0 += 4) {
    int ka = k0 + kh;
    v2f a, bv;
    const float* p0 = (ka < HH) ? (seq + m * HH + ka) : (spans + m * LL + (ka - HH));
    const float* p1 = (ka + 1 < HH) ? (seq + m * HH + ka + 1) : (spans + m * LL + (ka + 1 - HH));
    a.x = *p0;
    a.y = *p1;
    bv.x = W_pol[ka * LL + nc] * bmask;
    bv.y = W_pol[(ka + 1) * LL + nc] * bmask;
    acc = __builtin_amdgcn_wmma_f32_16x16x4_f32(false, a, false, bv, (short)0, acc, false, false);
  }
  if (n < LL) {
#pragma unroll
    for (int r = 0; r < 8; ++r) {
      int row = mt * 16 + r + 8 * (lane >> 4);
      pol_out[row * LL + n] = acc[r] + b_pol[n];
    }
  }
}

extern "C" void kernel_launch(void* const* d_in, const int* in_sizes, int n_in,
                              void* d_out, int out_size, void* d_ws, size_t ws_size,
                              hipStream_t stream) {
  const float* tok     = (const float*)d_in[0];
  const float* seq     = (const float*)d_in[1];
  const float* w_start = (const float*)d_in[2];
  const float* b_start = (const float*)d_in[3];
  const float* w_end   = (const float*)d_in[4];
  const float* b_end   = (const float*)d_in[5];
  const float* W_nsp   = (const float*)d_in[6];
  const float* b_nsp   = (const float*)d_in[7];
  const float* W_span  = (const float*)d_in[8];
  const float* b_span  = (const float*)d_in[9];
  const float* W_pol   = (const float*)d_in[10];
  const float* b_pol   = (const float*)d_in[11];

  float* out = (float*)d_out;
  float* start_logits = out;                       // [64,512]
  float* end_logits   = out + BB * SS;             // [64,512]
  float* spans        = out + 2 * BB * SS;         // [64,20]
  float* nsp_out      = spans + BB * LL;           // [64,9]
  float* pol_out      = nsp_out + BB * NSP;        // [64,20]
  float* starts_oh    = pol_out + BB * LL;         // [64,512]
  float* ends_oh      = starts_oh + BB * SS;       // [64,512]
  float* labels_oh    = ends_oh + BB * SS;         // [64,20]

  int* ws_i = (int*)d_ws;

  // 1) fused start/end logits: the single bandwidth-dominant pass (~100 MB),
  //    4 rows per wave with register-blocked weights (token loads dominate VMEM issue)
  k_row_logits<<<(BB * SS) / 32, 256, 0, stream>>>(tok, w_start, w_end, b_start, b_end,
                                                   start_logits, end_logits);
  // 2) seq-level logits; stage full_seq_logits in pol region (overwritten by k_pol later)
  k_seq<<<BB, 256, 0, stream>>>(seq, W_span, b_span, W_nsp, b_nsp, pol_out, nsp_out, ws_i);
  // 3) per-item top-8 + faithful span-selection loop
  k_select<<<BB, 32, 0, stream>>>(start_logits, end_logits, ws_i);
  // 4) span pooling (L2-resident) + WMMA span classifier + one-hot outputs
  k_span_pool<<<BB, 256, 0, stream>>>(tok, W_span, b_span, pol_out, ws_i,
                                      spans, starts_oh, ends_oh, labels_oh);
  // 5) pol_logits GEMM via WMMA (overwrites the staged fsl)
  k_pol<<<1, 256, 0, stream>>>(seq, spans, W_pol, b_pol, pol_out);
}